// lstm_model_1486058684494
// MI455X (gfx1250) — compile-verified
//
#include <hip/hip_runtime.h>

#define B_SZ 4096
#define T_SZ 2048
#define I_SZ 5

typedef __attribute__((ext_vector_type(2))) float v2f;
typedef __attribute__((ext_vector_type(8))) float v8f;

// ---------------- fast activations (prefer gfx1250 V_TANH_F32) ----------------
__device__ __forceinline__ float hw_tanh(float x) {
#if __has_builtin(__builtin_amdgcn_tanhf)
    return __builtin_amdgcn_tanhf(x);
#elif __has_builtin(__builtin_amdgcn_tanh_f32)
    return __builtin_amdgcn_tanh_f32(x);
#else
    // tanh(x) = 2/(1+e^{-2x}) - 1 ; e^{-2x} = exp2(-2*log2(e)*x)
    float e = __builtin_amdgcn_exp2f(x * -2.885390081777927f);
    return __builtin_fmaf(2.0f, __builtin_amdgcn_rcpf(1.0f + e), -1.0f);
#endif
}

__device__ __forceinline__ float fast_sigmoid(float x) {
#if __has_builtin(__builtin_amdgcn_tanhf) || __has_builtin(__builtin_amdgcn_tanh_f32)
    return __builtin_fmaf(0.5f, hw_tanh(0.5f * x), 0.5f);
#else
    float e = __builtin_amdgcn_exp2f(x * -1.4426950408889634f);
    return __builtin_amdgcn_rcpf(1.0f + e);
#endif
}

// ---------------- Phase 1: gate pre-activations via WMMA 16x16x4 f32 ----------
// D = A(16x4) x B(4x16) + C, wave32.
//   A rows m = gate (m<4 valid): A[m][k] = W_ih[m][k]          (k = 0..3)
//   B cols n = 16 consecutive time rows: B[k][n] = x[row n][k]
//   C = broadcast (b_ih+b_hh) per gate row
//   k=4 term folded post-WMMA: D[g][n] += x[row n][4]*W_ih[g][4]
// A layout (ISA 7.12.2): lane L, m=L&15; v0 = K=(half?2:0), v1 = K=(half?3:1)
// B layout: v0 = rows K=0 (lanes 0-15) / K=2 (lanes 16-31); v1 = K=1 / K=3
// D layout: VGPR g, lanes 0-15 = gate g over the 16 rows (lanes 16-31 garbage)
__global__ __launch_bounds__(256) void lstm_gates_wmma(
    const float* __restrict__ x, const float* __restrict__ W_ih,
    const float* __restrict__ b_ih, const float* __restrict__ b_hh,
    float4* __restrict__ xg, int t0, int tpbShift, int nTiles)
{
    const int lane = threadIdx.x & 31;
    const int n    = lane & 15;
    const int half = lane >> 4;
    const int waveId = (blockIdx.x * blockDim.x + threadIdx.x) >> 5;
    const int nWaves = (gridDim.x * blockDim.x) >> 5;

    // A-matrix (weights): loop-invariant
    const int mi = (n < 4) ? n : 0;
    float a0 = W_ih[mi * I_SZ + (half ? 2 : 0)];
    float a1 = W_ih[mi * I_SZ + (half ? 3 : 1)];
    if (n >= 4) { a0 = 0.0f; a1 = 0.0f; }
    v2f A; A[0] = a0; A[1] = a1;

    // C-matrix: bias per gate row (uniform per VGPR)
    const float bb0 = b_ih[0] + b_hh[0];
    const float bb1 = b_ih[1] + b_hh[1];
    const float bb2 = b_ih[2] + b_hh[2];
    const float bb3 = b_ih[3] + b_hh[3];
    v8f C; C[0]=bb0; C[1]=bb1; C[2]=bb2; C[3]=bb3; C[4]=0.f; C[5]=0.f; C[6]=0.f; C[7]=0.f;

    // 5th input column weights
    const float w40 = W_ih[0*I_SZ+4], w41 = W_ih[1*I_SZ+4];
    const float w42 = W_ih[2*I_SZ+4], w43 = W_ih[3*I_SZ+4];

    const int tpbMask = (1 << tpbShift) - 1;

    for (int tl = waveId; tl < nTiles; tl += nWaves) {
        const int b   = tl >> tpbShift;             // batch element
        const int tt0 = (tl & tpbMask) << 4;        // time offset within chunk
        const int tt  = tt0 + n;                    // this lane's row
        const long gr = (long)b * T_SZ + t0 + tt;   // row in x [B,T,I]
        const float* xr = x + gr * I_SZ;

        v2f Bm; Bm[0] = xr[half ? 2 : 0]; Bm[1] = xr[half ? 3 : 1];
        const float x4 = xr[4];

        v8f D = __builtin_amdgcn_wmma_f32_16x16x4_f32(
            false, A, false, Bm, (short)0, C, false, false);

        if (lane < 16) {
            float4 o;
            o.x = __builtin_fmaf(x4, w40, D[0]);
            o.y = __builtin_fmaf(x4, w41, D[1]);
            o.z = __builtin_fmaf(x4, w42, D[2]);
            o.w = __builtin_fmaf(x4, w43, D[3]);
            xg[(long)tt * B_SZ + b] = o;   // time-major [tt][b] float4
        }
    }
}

// ---------------- Phase 2: sequential LSTM scan (latency-bound) ---------------
__global__ __launch_bounds__(256) void lstm_scan(
    const float4* __restrict__ xg, const float* __restrict__ W_hh,
    const float* __restrict__ W_out, const float* __restrict__ b_out,
    float* __restrict__ out, float* __restrict__ state,
    int t0, int chunkT, int zeroInit)
{
    const int b = blockIdx.x * blockDim.x + threadIdx.x;
    if (b >= B_SZ) return;
    const float whi = W_hh[0], whf = W_hh[1], whg = W_hh[2], who = W_hh[3];
    const float wy  = W_out[0], by = b_out[0];

    float h, c;
    if (zeroInit) { h = 0.0f; c = 0.0f; }
    else          { h = state[b]; c = state[B_SZ + b]; }

    float* __restrict__ yp = out + (long)b * T_SZ + t0;
    const float4* __restrict__ gp = xg + b;

    #pragma unroll 4
    for (int t = 0; t < chunkT; ++t) {
        const float4 g4 = gp[(long)t * B_SZ];        // coalesced 512B/wave
        const float gi = __builtin_fmaf(h, whi, g4.x);
        const float gf = __builtin_fmaf(h, whf, g4.y);
        const float gg = __builtin_fmaf(h, whg, g4.z);
        const float go = __builtin_fmaf(h, who, g4.w);
        const float ia = fast_sigmoid(gi);
        const float fa = fast_sigmoid(gf);
        const float ga = hw_tanh(gg);
        const float oa = fast_sigmoid(go);
        c = __builtin_fmaf(fa, c, ia * ga);
        h = oa * hw_tanh(c);
        yp[t] = __builtin_fmaf(h, wy, by);
    }
    state[b] = h; state[B_SZ + b] = c;
}

// ---------------- Fallback: fully fused (used only if ws too small) -----------
__global__ __launch_bounds__(256) void lstm_fused(
    const float* __restrict__ x, const float* __restrict__ W_ih,
    const float* __restrict__ W_hh, const float* __restrict__ b_ih,
    const float* __restrict__ b_hh, const float* __restrict__ W_out,
    const float* __restrict__ b_out, float* __restrict__ out)
{
    const int b = blockIdx.x * blockDim.x + threadIdx.x;
    if (b >= B_SZ) return;
    float w[20];
    #pragma unroll
    for (int j = 0; j < 20; ++j) w[j] = W_ih[j];
    const float bi = b_ih[0] + b_hh[0], bf = b_ih[1] + b_hh[1];
    const float bg = b_ih[2] + b_hh[2], bo = b_ih[3] + b_hh[3];
    const float whi = W_hh[0], whf = W_hh[1], whg = W_hh[2], who = W_hh[3];
    const float wy  = W_out[0], by = b_out[0];

    const float* xp = x + (long)b * T_SZ * I_SZ;
    float* yp = out + (long)b * T_SZ;
    float h = 0.0f, c = 0.0f;
    for (int t = 0; t < T_SZ; ++t) {
        const float x0 = xp[0], x1 = xp[1], x2 = xp[2], x3 = xp[3], x4 = xp[4];
        xp += I_SZ;
        float gi = __builtin_fmaf(x4,w[4], __builtin_fmaf(x3,w[3], __builtin_fmaf(x2,w[2], __builtin_fmaf(x1,w[1], __builtin_fmaf(x0,w[0], __builtin_fmaf(h,whi,bi))))));
        float gf = __builtin_fmaf(x4,w[9], __builtin_fmaf(x3,w[8], __builtin_fmaf(x2,w[7], __builtin_fmaf(x1,w[6], __builtin_fmaf(x0,w[5], __builtin_fmaf(h,whf,bf))))));
        float gg = __builtin_fmaf(x4,w[14],__builtin_fmaf(x3,w[13],__builtin_fmaf(x2,w[12],__builtin_fmaf(x1,w[11],__builtin_fmaf(x0,w[10],__builtin_fmaf(h,whg,bg))))));
        float go = __builtin_fmaf(x4,w[19],__builtin_fmaf(x3,w[18],__builtin_fmaf(x2,w[17],__builtin_fmaf(x1,w[16],__builtin_fmaf(x0,w[15],__builtin_fmaf(h,who,bo))))));
        const float ia = fast_sigmoid(gi);
        const float fa = fast_sigmoid(gf);
        const float ga = hw_tanh(gg);
        const float oa = fast_sigmoid(go);
        c = __builtin_fmaf(fa, c, ia * ga);
        h = oa * hw_tanh(c);
        yp[t] = __builtin_fmaf(h, wy, by);
    }
}

extern "C" void kernel_launch(void* const* d_in, const int* in_sizes, int n_in,
                              void* d_out, int out_size, void* d_ws, size_t ws_size,
                              hipStream_t stream)
{
    (void)in_sizes; (void)n_in; (void)out_size;
    const float* x     = (const float*)d_in[0];
    const float* W_ih  = (const float*)d_in[1];
    const float* W_hh  = (const float*)d_in[2];
    const float* b_ih  = (const float*)d_in[3];
    const float* b_hh  = (const float*)d_in[4];
    const float* W_out = (const float*)d_in[5];
    const float* b_out = (const float*)d_in[6];
    float* out = (float*)d_out;

    const size_t stateBytes = (size_t)2 * B_SZ * sizeof(float);
    int chunkT = 0;
    for (int ct = T_SZ; ct >= 16; ct >>= 1) {
        if (stateBytes + (size_t)ct * B_SZ * sizeof(float4) <= ws_size) { chunkT = ct; break; }
    }

    if (chunkT == 0) {
        lstm_fused<<<(B_SZ + 255) / 256, 256, 0, stream>>>(
            x, W_ih, W_hh, b_ih, b_hh, W_out, b_out, out);
        return;
    }

    float*  state = (float*)d_ws;
    float4* xg    = (float4*)((char*)d_ws + stateBytes);

    const int tilesPerB = chunkT >> 4;
    int tpbShift = 0;
    while ((1 << tpbShift) < tilesPerB) ++tpbShift;
    const int nTiles = B_SZ * tilesPerB;

    for (int t0 = 0; t0 < T_SZ; t0 += chunkT) {
        lstm_gates_wmma<<<512, 256, 0, stream>>>(x, W_ih, b_ih, b_hh, xg, t0, tpbShift, nTiles);
        lstm_scan<<<(B_SZ + 255) / 256, 256, 0, stream>>>(
            xg, W_hh, W_out, b_out, out, state, t0, chunkT, t0 == 0 ? 1 : 0);
    }
}